// RMacPool2d_28467043238140
// MI455X (gfx1250) — compile-verified
//
#include <hip/hip_runtime.h>
#include <hip/hip_bf16.h>
#include <cstdint>

// RMAC pooling for x:(32,2048,32,32) f32 -> out:(32,2048) f32.
// Memory-bound: 256MB read ~= 11us at 23.3 TB/s. CDNA5 path used:
// global_load_async_to_lds_b128 (ASYNCcnt) bulk staging with NT hint
// (single-pass stream, keep L2 for the ws intermediate) + s_wait_asynccnt,
// wave32 shuffle reductions. No matmul -> WMMA not applicable.

#define WAVES_PER_BLOCK 8
#define NEG_BIG (-3.402823466e38f)

// 8 distinct [start,end) intervals for rows and columns (H=W=32, L=3):
// k: 0=[0,32) 1=[0,21) 2=[6,27) 3=[11,32) 4=[0,16) 5=[5,21) 6=[11,27) 7=[16,32)
__device__ __constant__ const int kRS[8] = {0, 0, 6, 11, 0, 5, 11, 16};
__device__ __constant__ const int kRE[8] = {32, 21, 27, 32, 16, 21, 27, 32};
// 26 unique regions (region 0 carries weight 4 = the four identical l=1 regions)
__device__ __constant__ const int kRR[26] = {0, 1,1,1, 2,2,2, 3,3,3,
                                             4,4,4,4, 5,5,5,5, 6,6,6,6, 7,7,7,7};
__device__ __constant__ const int kCC[26] = {0, 1,2,3, 1,2,3, 1,2,3,
                                             4,5,6,7, 4,5,6,7, 4,5,6,7, 4,5,6,7};

__global__ __launch_bounds__(256) void rmac_phaseA(const float* __restrict__ x,
                                                   float* __restrict__ ws) {
  __shared__ float tile[WAVES_PER_BLOCK][1024];  // 4KB plane per wave
  const int lane = threadIdx.x & 31;
  const int wv   = threadIdx.x >> 5;
  const unsigned plane = blockIdx.x * WAVES_PER_BLOCK + wv;  // 0..65535 exact

  // ---- Stage the 4KB (32x32 f32) plane into LDS via async copy engine.
  // NT temporal hint: x is a single-pass 256MB stream; don't pollute L2.
  const float* src = x + (size_t)plane * 1024u;
  const uint32_t ldsBase = (uint32_t)(uintptr_t)(&tile[wv][0]);
#pragma unroll
  for (int ch = 0; ch < 8; ++ch) {
    uint64_t ga = (uint64_t)(uintptr_t)(src + ch * 128 + lane * 4);  // 16B/lane
    uint32_t la = ldsBase + (uint32_t)(ch * 512 + lane * 16);
    asm volatile("global_load_async_to_lds_b128 %0, %1, off th:TH_LOAD_NT"
                 :: "v"(la), "v"(ga) : "memory");
  }
  asm volatile("s_wait_asynccnt 0" ::: "memory");

  // ---- lane = column; stream 32 rows, keep 8 row-range running maxes ----
  float rmax[8];
#pragma unroll
  for (int k = 0; k < 8; ++k) rmax[k] = NEG_BIG;
  const float* trow = &tile[wv][0];
#pragma unroll
  for (int row = 0; row < 32; ++row) {
    float v = trow[row * 32 + lane];  // lane-consecutive -> conflict-free
#pragma unroll
    for (int k = 0; k < 8; ++k)
      if (row >= kRS[k] && row < kRE[k]) rmax[k] = fmaxf(rmax[k], v);
  }

  // ---- 26 masked wave32 butterfly max reductions over column ranges ----
  float outv = 0.0f;
#pragma unroll
  for (int r = 0; r < 26; ++r) {
    const int cc = kCC[r];
    float v = (lane >= kRS[cc] && lane < kRE[cc]) ? rmax[kRR[r]] : NEG_BIG;
#pragma unroll
    for (int off = 16; off > 0; off >>= 1)
      v = fmaxf(v, __shfl_xor(v, off, 32));
    if (lane == r) outv = v;  // gather region r's result onto lane r
  }
  if (lane < 26) ws[(size_t)plane * 32u + lane] = outv;  // coalesced
}

__global__ __launch_bounds__(256) void rmac_phaseB(const float* __restrict__ ws,
                                                   float* __restrict__ out) {
  const int b    = blockIdx.x;      // 32 blocks, one batch each
  const int tid  = threadIdx.x;     // 256 threads -> 8 channels/thread
  const int lane = tid & 31;
  const int wv   = tid >> 5;

  __shared__ float wsum[8][26];
  __shared__ float scl[26];
  __shared__ float wsum2[8];
  __shared__ float fscale;

  // ---- Pass 1: per-region sum of squares over C (deterministic order) ----
  float ss[26];
#pragma unroll
  for (int r = 0; r < 26; ++r) ss[r] = 0.0f;
#pragma unroll
  for (int k = 0; k < 8; ++k) {
    const float* p = ws + ((size_t)(b * 2048 + tid + k * 256) << 5);
#pragma unroll
    for (int r = 0; r < 26; ++r) { float t = p[r]; ss[r] += t * t; }
  }
#pragma unroll
  for (int r = 0; r < 26; ++r) {
#pragma unroll
    for (int off = 16; off > 0; off >>= 1)
      ss[r] += __shfl_xor(ss[r], off, 32);
  }
  if (lane == 0) {
#pragma unroll
    for (int r = 0; r < 26; ++r) wsum[wv][r] = ss[r];
  }
  __syncthreads();
  if (tid < 26) {
    float tot = 0.0f;
    for (int w = 0; w < 8; ++w) tot += wsum[w][tid];
    const float wgt = (tid == 0) ? 4.0f : 1.0f;  // 4 identical l=1 regions
    scl[tid] = wgt / (sqrtf(tot) + 1e-6f);
  }
  __syncthreads();

  // ---- Pass 2: weighted accumulate + final norm ----
  float acc[8];
  float s2 = 0.0f;
#pragma unroll
  for (int k = 0; k < 8; ++k) {
    const float* p = ws + ((size_t)(b * 2048 + tid + k * 256) << 5);
    float a = 0.0f;
#pragma unroll
    for (int r = 0; r < 26; ++r) a += scl[r] * p[r];
    acc[k] = a;
    s2 += a * a;
  }
#pragma unroll
  for (int off = 16; off > 0; off >>= 1) s2 += __shfl_xor(s2, off, 32);
  if (lane == 0) wsum2[wv] = s2;
  __syncthreads();
  if (tid == 0) {
    float tot = 0.0f;
    for (int w = 0; w < 8; ++w) tot += wsum2[w];
    fscale = 1.0f / (sqrtf(tot) + 1e-6f);
  }
  __syncthreads();
  const float fs = fscale;
#pragma unroll
  for (int k = 0; k < 8; ++k)
    out[b * 2048 + tid + k * 256] = acc[k] * fs;
}

extern "C" void kernel_launch(void* const* d_in, const int* in_sizes, int n_in,
                              void* d_out, int out_size, void* d_ws, size_t ws_size,
                              hipStream_t stream) {
  const float* x = (const float*)d_in[0];        // (32,2048,32,32) f32
  float* out = (float*)d_out;                    // (32,2048) f32
  float* ws = (float*)d_ws;                      // needs 65536*32*4 = 8 MB

  // 65536 planes, 8 planes (waves) per block
  rmac_phaseA<<<65536 / WAVES_PER_BLOCK, 256, 0, stream>>>(x, ws);
  rmac_phaseB<<<32, 256, 0, stream>>>(ws, out);
}